// Get_Cat_Feat_Tgt_45999099740712
// MI455X (gfx1250) — compile-verified
//
#include <hip/hip_runtime.h>
#include <cstdint>

// ---------------------------------------------------------------------------
// Problem constants (from reference setup_inputs)
#define B_    2
#define KTOP  128
#define C_    10
#define S_    (KTOP * C_)     // 1280 query points per batch
#define N_    16384           // reference points per batch
#define F_    32              // feature channels
#define KNN   32              // nsample
#define CH    (3 + F_)        // output channels = 35
#define R2    4.0f            // radius^2

typedef __attribute__((ext_vector_type(2))) float v2f;
typedef __attribute__((ext_vector_type(8))) float v8f;

__device__ __forceinline__ uint32_t ballot32(bool p) {
#if __has_builtin(__builtin_amdgcn_ballot_w32)
    return __builtin_amdgcn_ballot_w32(p);
#else
    return (uint32_t)__ballot(p);
#endif
}

// ---------------------------------------------------------------------------
// Kernel 1: WMMA-based ball query.
// One wave (32 lanes) per tile of 16 query points.  Per iteration it evaluates
// dval = -2 q.x + ||x||^2 for a 16x32 (queries x points) block via two
// V_WMMA_F32_16X16X4_F32 ops, ballots the in-radius predicate, and lanes 0..15
// pop set bits (ascending index order == reference's sort-then-truncate) into
// an LDS ring until each row has KNN hits.  Uniform early exit via ballot.
template<bool FUSED>
__global__ __launch_bounds__(32)
void ball_query_wmma_k(const float* __restrict__ cand,   // (B, S, 3) queries
                       const float* __restrict__ xyz,    // (B, N, 3)
                       const float* __restrict__ feat,   // (B, N, F)
                       int*   __restrict__ nn_idx,       // (B*S, KNN)   [!FUSED]
                       float* __restrict__ out)          // (B*S,KNN,CH) [FUSED]
{
    __shared__ int lds_idx[16 * KNN];

    const int lane  = threadIdx.x;            // 0..31
    const int tile  = blockIdx.x;             // B_*S_/16 tiles
    const int qbase = tile * 16;              // flattened query index base
    const int b     = qbase / S_;             // S_ % 16 == 0 -> no straddle
    const float* xyzb = xyz + (size_t)b * N_ * 3;

    // ---- load this lane's query row (row = lane & 15) --------------------
    const int   row = lane & 15;
    const float qx  = cand[(qbase + row) * 3 + 0];
    const float qy  = cand[(qbase + row) * 3 + 1];
    const float qz  = cand[(qbase + row) * 3 + 2];
    // include iff  dval <= r^2 - ||q||^2   (matches reference's  sqr > r^2  exclusion)
    const float thr_self = R2 - (qx * qx + qy * qy + qz * qz);

    // ---- A fragment (16x4 f32): lanes 0-15 hold K=0,1; lanes 16-31 K=2,3 --
    v2f a;
    if (lane < 16) { a.x = qx; a.y = qy; }
    else           { a.x = qz; a.y = 1.0f; }

    // ---- per-lane thresholds for D rows: VGPR v = row v (lanes<16) / v+8 --
    float thr[8];
    const int rbase = (lane < 16) ? 0 : 8;
#pragma unroll
    for (int v = 0; v < 8; ++v)
        thr[v] = __shfl(thr_self, rbase + v, 32);

    int count = (lane < 16) ? 0 : KNN;        // lanes >= 16 never extract
    const int ncol = lane & 15;

    for (int base_n = 0; base_n < N_; base_n += 32) {
        // B fragments for columns [base_n, base_n+16) and [base_n+16, base_n+32)
        v2f bm0, bm1;
        {
            const float* p0 = xyzb + (size_t)(base_n + ncol) * 3;
            const float* p1 = p0 + 16 * 3;
            float x0 = p0[0], y0 = p0[1], z0 = p0[2];
            float x1 = p1[0], y1 = p1[1], z1 = p1[2];
            if (lane < 16) {
                bm0.x = -2.0f * x0; bm0.y = -2.0f * y0;
                bm1.x = -2.0f * x1; bm1.y = -2.0f * y1;
            } else {
                bm0.x = -2.0f * z0; bm0.y = x0 * x0 + y0 * y0 + z0 * z0;
                bm1.x = -2.0f * z1; bm1.y = x1 * x1 + y1 * y1 + z1 * z1;
            }
        }

        v8f c0 = {0.f, 0.f, 0.f, 0.f, 0.f, 0.f, 0.f, 0.f};
        v8f d0 = __builtin_amdgcn_wmma_f32_16x16x4_f32(false, a, false, bm0,
                                                       (short)0, c0, false, false);
        v8f d1 = __builtin_amdgcn_wmma_f32_16x16x4_f32(false, a, false, bm1,
                                                       (short)0, c0, false, false);

        uint32_t m0[8], m1[8];
#pragma unroll
        for (int v = 0; v < 8; ++v) {
            m0[v] = ballot32(d0[v] <= thr[v]);   // bits[15:0]=row v, [31:16]=row v+8
            m1[v] = ballot32(d1[v] <= thr[v]);
        }

        // lane r (<16) extracts its row's 32-bit column mask and pops bits
        if (lane < 16) {
            uint32_t s0 = m0[0], s1 = m1[0];
#pragma unroll
            for (int v = 1; v < 8; ++v) {
                bool sel = ((lane & 7) == v);
                s0 = sel ? m0[v] : s0;
                s1 = sel ? m1[v] : s1;
            }
            uint32_t b0 = (lane < 8) ? (s0 & 0xFFFFu) : (s0 >> 16);
            uint32_t b1 = (lane < 8) ? (s1 & 0xFFFFu) : (s1 >> 16);
            uint32_t bits = b0 | (b1 << 16);      // bit k -> column base_n + k
            while (bits && count < KNN) {
                int bit = __ffs(bits) - 1;
                lds_idx[(lane << 5) | count] = base_n + bit;
                ++count;
                bits &= bits - 1;
            }
        }
        if (ballot32(count >= KNN) == 0xFFFFFFFFu) break;   // uniform early exit
    }

    __syncthreads();
    // pad: repeat first hit; empty row -> N_ (gather clamps like jnp OOB index)
    if (lane < 16) {
        int first = (count > 0) ? lds_idx[lane << 5] : N_;
        for (int k = count; k < KNN; ++k) lds_idx[(lane << 5) | k] = first;
    }
    __syncthreads();

    if (!FUSED) {
        for (int i = lane; i < 16 * KNN; i += 32)
            nn_idx[qbase * KNN + i] = lds_idx[i];
    } else {
        const float* featb = feat + (size_t)b * N_ * F_;
        for (int pair = 0; pair < 16 * KNN; ++pair) {
            int idx = lds_idx[pair];
            if (idx >= N_) idx = N_ - 1;          // jnp clamp semantics
            float* o = out + ((size_t)qbase * KNN + pair) * CH;
            o[3 + lane] = featb[(size_t)idx * F_ + lane];
            if (lane < 3) o[lane] = xyzb[(size_t)idx * 3 + lane] * 0.5f;
        }
    }
}

// ---------------------------------------------------------------------------
// Kernel 2: fully parallel gather + concat; one thread per output element.
// Consecutive threads cover consecutive channels -> coalesced feature traffic.
__global__ __launch_bounds__(256)
void gather_concat_k(const float* __restrict__ xyz,
                     const float* __restrict__ feat,
                     const int*   __restrict__ nn_idx,
                     float* __restrict__ out, int total)
{
    int i = blockIdx.x * blockDim.x + threadIdx.x;
    if (i >= total) return;
    int ch   = i % CH;
    int pair = i / CH;            // (b*S + s)*KNN + k
    int q    = pair / KNN;        // b*S + s
    int b    = q / S_;
    int idx  = nn_idx[pair];
    if (idx >= N_) idx = N_ - 1;  // jnp clamp semantics
    float v;
    if (ch < 3) v = xyz [((size_t)b * N_ + idx) * 3  + ch]        * 0.5f;
    else        v = feat[((size_t)b * N_ + idx) * F_ + (ch - 3)];
    out[i] = v;
}

// ---------------------------------------------------------------------------
extern "C" void kernel_launch(void* const* d_in, const int* in_sizes, int n_in,
                              void* d_out, int out_size, void* d_ws, size_t ws_size,
                              hipStream_t stream)
{
    (void)in_sizes; (void)n_in; (void)out_size;
    const float* cand = (const float*)d_in[0];   // candidate_pts (B,Ktop,C,3) == (B,S,3)
    // d_in[1] = src_keypts: unused by the reference computation
    const float* xyz  = (const float*)d_in[2];   // tgt_pts_xyz (B,N,3)
    const float* feat = (const float*)d_in[3];   // tgt_deep_feat_pts (B,N,F)
    float* out = (float*)d_out;

    const int tiles = (B_ * S_) / 16;            // 160 query tiles
    const size_t idx_bytes = (size_t)B_ * S_ * KNN * sizeof(int);

    if (ws_size >= idx_bytes) {
        int* nn = (int*)d_ws;
        ball_query_wmma_k<false><<<tiles, 32, 0, stream>>>(cand, xyz, feat, nn, nullptr);
        const int total = B_ * S_ * KNN * CH;    // 2,867,200
        gather_concat_k<<<(total + 255) / 256, 256, 0, stream>>>(xyz, feat, nn, out, total);
    } else {
        // scratch too small: fused in-wave gather (no workspace needed)
        ball_query_wmma_k<true><<<tiles, 32, 0, stream>>>(cand, xyz, feat, nullptr, out);
    }
}